// FixedRandomMap_alt_51951924413085
// MI455X (gfx1250) — compile-verified
//
#include <hip/hip_runtime.h>

// ---------------------------------------------------------------------------
// out[i] = T[k_i-1] @ ... @ T[0] @ x[i],  k_i = fwd_steps[i]  (k_i==0 -> x[i])
//
// Kogge-Stone parallel scan over the 64 chained matrices (6 levels of
// independent 256^3 f32 matmuls) + one batched apply. Entire working set
// (x 4MB + T 16MB + 32MB ping-pong) is L2-resident on MI455X (192MB L2),
// so the problem is latency/launch bound: scan depth 6 instead of 64.
// f32 WMMA (v_wmma_f32_16x16x4_f32) keeps reference f32 numerics.
//
// Each wave32 computes a 32x32 output tile: 2x2 register blocking over
// 16x16 WMMA fragments -> 4 WMMAs per (2 b64 + 4 b32) loads, with all inner
// loads/stores folded to base+immediate offsets via partial unroll.
// ---------------------------------------------------------------------------

typedef float v2f __attribute__((ext_vector_type(2)));
typedef float v8f __attribute__((ext_vector_type(8)));

#define SDIM   256   // matrix side
#define KSTEPS 64    // number of T matrices
#define BATCH  16    // samples

#define WMMA_F32(A, B, C) \
  __builtin_amdgcn_wmma_f32_16x16x4_f32(false, (A), false, (B), (short)0, (C), false, false)

// One wave32: D[m0:m0+32, n0:n0+32] = A(32 x S) * B(S x 32).
// Fragment layouts per CDNA5 ISA 7.12.2 (f32, 16x16x4):
//   A 16x4 : lane L holds A[L%16][2*(L/16) + {0,1}]  (contiguous pair -> b64)
//   B 4x16 : lane L holds B[2*(L/16) + {0,1}][L%16]
//   C/D    : lane L, vgpr r holds D[r + 8*(L/16)][L%16]
__device__ __forceinline__ void mm_tile_32x32(const float* __restrict__ A,
                                              const float* __restrict__ Bm,
                                              float* __restrict__ D,
                                              int m0, int n0, int lane) {
  const int mr = lane & 15;  // row (A) / col (B,D) within a fragment
  const int lh = lane >> 4;  // lane-high selector

  const float* pa = A  + (size_t)(m0 + mr) * SDIM + 2 * lh;   // advance by k
  const float* pb = Bm + (size_t)(2 * lh) * SDIM + n0 + mr;   // advance by k*S

  v8f acc00 = {}, acc01 = {}, acc10 = {}, acc11 = {};

#pragma unroll 8
  for (int kk = 0; kk < SDIM; kk += 4) {
    v2f a0 = *(const v2f*)(pa);                 // rows m0..m0+15
    v2f a1 = *(const v2f*)(pa + 16 * SDIM);     // rows m0+16..m0+31
    v2f b0, b1;
    b0.x = pb[0];           b0.y = pb[SDIM];        // cols n0..n0+15
    b1.x = pb[16];          b1.y = pb[SDIM + 16];   // cols n0+16..n0+31
    acc00 = WMMA_F32(a0, b0, acc00);
    acc01 = WMMA_F32(a0, b1, acc01);
    acc10 = WMMA_F32(a1, b0, acc10);
    acc11 = WMMA_F32(a1, b1, acc11);
    pa += 4;
    pb += 4 * SDIM;
  }

  float* d = D + (size_t)(m0 + 8 * lh) * SDIM + n0 + mr;
#pragma unroll
  for (int r = 0; r < 8; ++r) {
    d[(size_t)r * SDIM]             = acc00[r];
    d[(size_t)r * SDIM + 16]        = acc01[r];
    d[(size_t)(r + 16) * SDIM]      = acc10[r];
    d[(size_t)(r + 16) * SDIM + 16] = acc11[r];
  }
}

// Wave-uniform 32x32 tile copy: one row (32 floats = 8 float4) per lane.
__device__ __forceinline__ void copy_tile_32x32(const float* __restrict__ src,
                                                float* __restrict__ dst,
                                                int m0, int n0, int lane) {
  const float* s = src + (size_t)(m0 + lane) * SDIM + n0;
  float*       d = dst + (size_t)(m0 + lane) * SDIM + n0;
#pragma unroll
  for (int c = 0; c < 32; c += 4) {
    *(float4*)(d + c) = *(const float4*)(s + c);
  }
}

// Scan level: dst[j] = src[j] * src[j-shift] for j >= shift, else copy src[j].
// 64 matrices * 64 tiles = 4096 waves; 8 waves / 256-thread block.
__global__ void __launch_bounds__(256)
scan_level_kernel(const float* __restrict__ src, float* __restrict__ dst,
                  int shift) {
  const int lane = threadIdx.x & 31;
  const int wid  = blockIdx.x * 8 + (threadIdx.x >> 5);
  const int j    = wid >> 6;          // which matrix
  const int t    = wid & 63;          // 32x32 tile within matrix
  const int m0   = (t >> 3) << 5;
  const int n0   = (t & 7) << 5;
  const float* left = src + (size_t)j * SDIM * SDIM;
  float*       out  = dst + (size_t)j * SDIM * SDIM;
  if (j < shift) {                    // wave-uniform branch: EXEC stays full
    copy_tile_32x32(left, out, m0, n0, lane);
  } else {
    const float* right = src + (size_t)(j - shift) * SDIM * SDIM;
    mm_tile_32x32(left, right, out, m0, n0, lane);  // newer factor on the left
  }
}

// Apply: out[i] = (k==0) ? x[i] : prefix[k-1] * x[i]
__global__ void __launch_bounds__(256)
apply_kernel(const float* __restrict__ x, const float* __restrict__ prefix,
             const int* __restrict__ steps, float* __restrict__ out) {
  const int lane = threadIdx.x & 31;
  const int wid  = blockIdx.x * 8 + (threadIdx.x >> 5);
  const int i    = wid >> 6;          // sample
  const int t    = wid & 63;
  const int m0   = (t >> 3) << 5;
  const int n0   = (t & 7) << 5;
  const int k    = steps[i];          // wave-uniform
  const float* xi = x   + (size_t)i * SDIM * SDIM;
  float*       oi = out + (size_t)i * SDIM * SDIM;
  if (k == 0) {
    copy_tile_32x32(xi, oi, m0, n0, lane);
  } else {
    mm_tile_32x32(prefix + (size_t)(k - 1) * SDIM * SDIM, xi, oi, m0, n0, lane);
  }
}

extern "C" void kernel_launch(void* const* d_in, const int* in_sizes, int n_in,
                              void* d_out, int out_size, void* d_ws,
                              size_t ws_size, hipStream_t stream) {
  (void)in_sizes; (void)n_in; (void)out_size; (void)ws_size;
  const float* x     = (const float*)d_in[0];  // (16,256,256) f32
  const float* T     = (const float*)d_in[1];  // (64,256,256) f32 (binary)
  const int*   steps = (const int*)d_in[2];    // (16,) i32
  float*       out   = (float*)d_out;          // (16,256,256) f32

  const size_t matElems = (size_t)SDIM * SDIM;
  float* W0 = (float*)d_ws;                    // 16 MB
  float* W1 = W0 + (size_t)KSTEPS * matElems;  // 16 MB

  const dim3 blk(256);
  const int scanBlocks  = (KSTEPS * 64) / 8;   // 512 blocks
  const int applyBlocks = (BATCH * 64) / 8;    // 128 blocks

  // Kogge-Stone: after level with shift s, buf[j] = T[j]*T[j-1]*...*T[j-2s+1]
  const int shifts[6] = {1, 2, 4, 8, 16, 32};
  const float* cur = T;
  float* bufs[2] = {W0, W1};
  for (int l = 0; l < 6; ++l) {
    float* d = bufs[l & 1];
    scan_level_kernel<<<scanBlocks, blk, 0, stream>>>(cur, d, shifts[l]);
    cur = d;  // ends in W1 after 6 levels; src/dst never alias within a level
  }
  apply_kernel<<<applyBlocks, blk, 0, stream>>>(x, cur, steps, out);
}